// GCN_45870250721840
// MI455X (gfx1250) — compile-verified
//
#include <hip/hip_runtime.h>

typedef __attribute__((ext_vector_type(2))) float v2f;
typedef __attribute__((ext_vector_type(8))) float v8f;

#define NNODES 50000
#define NHID   256
#define NEDGE  800000
#define MT     4            // M-tiles (of 16 rows) per wave -> B-fragment reuse x4

// ---------------------------------------------------------------------------
// Zero-fill (float4 grid-stride)
// ---------------------------------------------------------------------------
__global__ void zero_kernel(float4* __restrict__ p, long n4) {
    long i = (long)blockIdx.x * blockDim.x + threadIdx.x;
    long stride = (long)gridDim.x * blockDim.x;
    float4 z; z.x = 0.f; z.y = 0.f; z.z = 0.f; z.w = 0.f;
    for (; i < n4; i += stride) p[i] = z;
}

// ---------------------------------------------------------------------------
// SpMM scatter-add: agg[dst[e], :] += w[e] * x[src[e], :]
// 4 edges per 256-thread block; 64 lanes per edge; float4 gather + f32 atomics
// ---------------------------------------------------------------------------
__global__ void spmm_kernel(const int* __restrict__ dst, const int* __restrict__ src,
                            const float* __restrict__ w, const float* __restrict__ X,
                            float* __restrict__ Y, int E) {
    int e = blockIdx.x * 4 + (threadIdx.x >> 6);
    if (e >= E) return;
    int t = threadIdx.x & 63;               // 64 lanes * 4 feats = 256
    int s = src[e];
    int d = dst[e];
    float we = w[e];
    const float4 xv = *(const float4*)(X + (long)s * NHID + t * 4);
    float* yp = Y + (long)d * NHID + t * 4;
    atomicAdd(yp + 0, we * xv.x);
    atomicAdd(yp + 1, we * xv.y);
    atomicAdd(yp + 2, we * xv.z);
    atomicAdd(yp + 3, we * xv.w);
}

// ---------------------------------------------------------------------------
// WMMA GEMM core: one wave computes NTILES vertically-stacked 16x16 tiles of
// out (a 16*NTILES x 16 column block), reusing each B fragment NTILES times.
// NTILES is compile-time and the dispatch is uniform per block, so EXEC is
// all-ones for every v_wmma (ISA requirement).
// ---------------------------------------------------------------------------
template <int K, int NTILES>
__device__ __forceinline__ void gemm_core(const float (*As)[K + 4],
                                          const float* __restrict__ W,
                                          const float* __restrict__ bias,
                                          float* __restrict__ out,
                                          long rowBase, int wave, int lane, int relu) {
    // Fragment layout (32-bit 16x4 A / 4x16 B, wave32):
    //   lanes 0-15 : VGPR0=K0, VGPR1=K1 ; lanes 16-31 : VGPR0=K2, VGPR1=K3
    const int arow  = lane & 15;
    const int khalf = (lane >> 4) * 2;
    const int col   = wave * 16 + (lane & 15);

    v8f c[NTILES];
    #pragma unroll
    for (int t = 0; t < NTILES; ++t) c[t] = {};

    #pragma unroll 2
    for (int k = 0; k < K; k += 4) {
        v2f b;
        b.x = W[(k + khalf) * NHID + col];
        b.y = W[(k + khalf + 1) * NHID + col];
        #pragma unroll
        for (int t = 0; t < NTILES; ++t) {
            v2f a;
            a.x = As[t * 16 + arow][k + khalf];
            a.y = As[t * 16 + arow][k + khalf + 1];
            c[t] = __builtin_amdgcn_wmma_f32_16x16x4_f32(
                       false, a, false, b, (short)0, c[t], false, false);
        }
    }

    // D layout: VGPR v -> row v (lanes 0-15) / v+8 (lanes 16-31), col = lane&15
    const float bv   = bias[col];
    const int   moff = (lane >> 4) * 8;
    #pragma unroll
    for (int t = 0; t < NTILES; ++t) {
        #pragma unroll
        for (int v = 0; v < 8; ++v) {
            float x = c[t][v] + bv;
            if (relu) x = fmaxf(x, 0.0f);
            out[(rowBase + t * 16 + v + moff) * NHID + col] = x;
        }
    }
}

// ---------------------------------------------------------------------------
// Fused GEMM + bias (+ReLU):  out[M,256] = act( A[M,K] @ W[K,256] + bias )
// Block = 512 threads = 16 waves; wave w owns column tile w; each wave stacks
// MT row tiles.  A strip (MT*16 rows x K) staged in LDS, padded +4 floats per
// row for conflict-free ds_load_b64 fragment reads.
// ---------------------------------------------------------------------------
template <int K>
__global__ void gemm_bias_wmma(const float* __restrict__ A, const float* __restrict__ W,
                               const float* __restrict__ bias, float* __restrict__ out,
                               int relu) {
    __shared__ float As[MT * 16][K + 4];
    const int tid  = threadIdx.x;           // 0..511
    const int wave = tid >> 5;              // 0..15 -> N tile index
    const int lane = tid & 31;
    const long rowBase = (long)blockIdx.x * (MT * 16);
    const int  rows    = min(MT * 16, NNODES - (int)rowBase);   // 64 or 16

    // Cooperative stage of the rows x K A-strip into LDS (float4, coalesced).
    for (int idx = tid; idx < rows * (K / 4); idx += 512) {
        int r  = idx / (K / 4);
        int kc = (idx % (K / 4)) * 4;
        float4 v = *(const float4*)(A + (rowBase + r) * K + kc);
        *(float4*)(&As[r][kc]) = v;
    }
    __syncthreads();

    // 3125 tiles = 4*781 + 1  ->  tilesLeft is 4 or 1, uniform per block.
    if (rows == MT * 16)
        gemm_core<K, MT>(As, W, bias, out, rowBase, wave, lane, relu);
    else
        gemm_core<K, 1>(As, W, bias, out, rowBase, wave, lane, relu);
}

// ---------------------------------------------------------------------------
// Head: out[M,C] = H[M,256] @ W2[256,C] + b2   (C in {2,6,21} -> plain VALU)
// ---------------------------------------------------------------------------
__global__ void head_kernel(const float* __restrict__ H, const float* __restrict__ W2,
                            const float* __restrict__ b2, float* __restrict__ out,
                            int M, int C) {
    int idx = blockIdx.x * blockDim.x + threadIdx.x;
    if (idx >= M * C) return;
    int row = idx / C;
    int cc  = idx % C;
    const float* h = H + (long)row * NHID;
    float acc = b2[cc];
    #pragma unroll 8
    for (int k = 0; k < NHID; ++k) acc += h[k] * W2[k * C + cc];
    out[idx] = acc;
}

// ---------------------------------------------------------------------------
extern "C" void kernel_launch(void* const* d_in, const int* in_sizes, int n_in,
                              void* d_out, int out_size, void* d_ws, size_t ws_size,
                              hipStream_t stream) {
    const float* features = (const float*)d_in[0];   // [50000,128]
    const int*   eidx     = (const int*)d_in[1];     // [2, 800000]
    const float* eweight  = (const float*)d_in[2];   // [800000]
    const float* W1       = (const float*)d_in[3];   // [128,256]
    const float* b1       = (const float*)d_in[4];   // [256]
    const float* Wg       = (const float*)d_in[5];   // [3,256,256]
    const float* bg       = (const float*)d_in[6];   // [3,256]
    const float* Wm1      = (const float*)d_in[7];   // [3,256,256]
    const float* bm1      = (const float*)d_in[8];   // [3,256]
    const float* Wm2[3]   = { (const float*)d_in[9],  (const float*)d_in[11], (const float*)d_in[13] };
    const float* bm2[3]   = { (const float*)d_in[10], (const float*)d_in[12], (const float*)d_in[14] };
    float* out = (float*)d_out;

    const int* dst = eidx;            // edge_index[0]
    const int* src = eidx + NEDGE;    // edge_index[1]

    const long actElems = (long)NNODES * NHID;          // 12.8M floats
    float* buf0 = (float*)d_ws;                         // current activation
    float* buf1 = buf0 + actElems;                      // spmm accumulator
    float* buf2 = buf1 + actElems;                      // head hidden

    const int mtiles = NNODES / 16;                     // 3125 (exact)
    const int gemmBlocks = (mtiles + MT - 1) / MT;      // 782
    const int ncls[3] = {2, 6, 21};
    const long outOff[3] = {0, (long)NNODES * 2, (long)NNODES * 8};

    // mlp1: buf0 = relu(features @ W1 + b1)
    gemm_bias_wmma<128><<<gemmBlocks, 512, 0, stream>>>(features, W1, b1, buf0, 1);

    for (int i = 0; i < 3; ++i) {
        // buf1 = 0 ; buf1[dst] += w * buf0[src]
        zero_kernel<<<4096, 256, 0, stream>>>((float4*)buf1, actElems / 4);
        spmm_kernel<<<(NEDGE + 3) / 4, 256, 0, stream>>>(dst, src, eweight, buf0, buf1, NEDGE);
        // buf0 = relu(buf1 @ Wg[i] + bg[i])
        gemm_bias_wmma<256><<<gemmBlocks, 512, 0, stream>>>(buf1, Wg + (long)i * NHID * NHID,
                                                            bg + (long)i * NHID, buf0, 1);
        // buf2 = relu(buf0 @ Wm1[i] + bm1[i])
        gemm_bias_wmma<256><<<gemmBlocks, 512, 0, stream>>>(buf0, Wm1 + (long)i * NHID * NHID,
                                                            bm1 + (long)i * NHID, buf2, 1);
        // out_i = buf2 @ Wm2[i] + bm2[i]
        int total = NNODES * ncls[i];
        head_kernel<<<(total + 255) / 256, 256, 0, stream>>>(buf2, Wm2[i], bm2[i],
                                                             out + outOff[i], NNODES, ncls[i]);
    }
}